// GAT_9431748182826
// MI455X (gfx1250) — compile-verified
//
#include <hip/hip_runtime.h>
#include <math.h>
#include <stdint.h>

#define NNODES   100000
#define NEDGES   1600000
#define ETOT     (NEDGES + NNODES)
#define HEADS    4
#define NEGSLOPE 0.2f

typedef float v2f __attribute__((ext_vector_type(2)));
typedef float v8f __attribute__((ext_vector_type(8)));
typedef unsigned int v4u __attribute__((ext_vector_type(4)));
typedef int v4i __attribute__((ext_vector_type(4)));
typedef int v8i __attribute__((ext_vector_type(8)));

// ---------------------------------------------------------------------------
// TDM: DMA a 128x128 f32 tile (row-major, stride 128) from global into LDS.
// Descriptor layout per CDNA5 ISA ch.8 (D# group0/group1).
// ---------------------------------------------------------------------------
__device__ __forceinline__ void tdm_load_w_to_lds(const float* W, float* lds_ptr) {
    // Flat LDS addresses carry the LDS byte offset in addr[31:0].
    const unsigned lds_addr = (unsigned)(uintptr_t)lds_ptr;
    const unsigned long long ga = (unsigned long long)(uintptr_t)W;

    v4u g0;
    g0.x = 1u;                                            // count=1 (valid), user desc
    g0.y = lds_addr;                                      // lds_addr[31:0]
    g0.z = (unsigned)(ga & 0xFFFFFFFFull);                // global_addr[31:0]
    g0.w = (unsigned)((ga >> 32) & 0x1FFFFFFull)          // global_addr[56:32]
           | (2u << 30);                                  // type = 2 ("image")

    v8i g1;
    g1.s0 = 0x00020000;        // workgroup_mask=0, data_size=2 (4B), no flags
    g1.s1 = (int)(128u << 16); // tensor_dim0[15:0]=128 in bits[63:48]
    g1.s2 = (int)(128u << 16); // tensor_dim1[15:0]=128 in bits[95:80]
    g1.s3 = (int)(128u << 16); // tile_dim0=128 in bits[127:112]
    g1.s4 = 128;               // tile_dim1=128 in bits[143:128], tile_dim2=0
    g1.s5 = 128;               // tensor_dim0_stride[31:0]=128
    g1.s6 = 0;                 // stride hi / tensor_dim1_stride lo
    g1.s7 = 0;

    v4i z4 = {0, 0, 0, 0};
#if __has_include(<hip/amd_detail/amd_gfx1250_TDM.h>)
    v8i z8 = {0, 0, 0, 0, 0, 0, 0, 0};
    __builtin_amdgcn_tensor_load_to_lds(g0, g1, z4, z4, z8, 0);   // clang-23 arity
#else
    __builtin_amdgcn_tensor_load_to_lds(g0, g1, z4, z4, 0);       // ROCm 7.2 arity
#endif
    __builtin_amdgcn_s_wait_tensorcnt(0);
}

// ---------------------------------------------------------------------------
// C[nrows,128] = (relu_in ? relu(A) : A)[nrows,128] @ W[128,128]
// WMMA f32 16x16x4, one wave per 16-row stripe, 8 column tiles per wave.
// ---------------------------------------------------------------------------
__global__ __launch_bounds__(256) void gemm128_wmma(const float* __restrict__ A,
                                                    const float* __restrict__ W,
                                                    float* __restrict__ C,
                                                    int nrows, int relu_in) {
    __shared__ float sW[128 * 128];   // 64 KB: whole weight matrix, k-major

    if (threadIdx.x < 32)             // wave 0 issues the tensor DMA + waits
        tdm_load_w_to_lds(W, sW);
    __syncthreads();                  // LDS valid for all 8 waves

    const int tid  = threadIdx.x;
    const int wave = tid >> 5;
    const int lane = tid & 31;
    const int m0 = blockIdx.x * 128 + wave * 16;
    if (m0 >= nrows) return;          // wave-uniform: EXEC stays all-ones below

    const int lrow  = lane & 15;            // row-in-tile / col-in-tile
    const int khalf = (lane >> 4) << 1;     // 0 for lanes 0-15, 2 for 16-31

    v8f acc[8];
#pragma unroll
    for (int t = 0; t < 8; ++t) acc[t] = (v8f)0.0f;

    const float* arow = A + (size_t)(m0 + lrow) * 128;

    for (int k0 = 0; k0 < 128; k0 += 4) {
        const int kk = k0 + khalf;
        v2f a;
        a.x = arow[kk];
        a.y = arow[kk + 1];
        if (relu_in) { a.x = fmaxf(a.x, 0.0f); a.y = fmaxf(a.y, 0.0f); }
#pragma unroll
        for (int t = 0; t < 8; ++t) {
            v2f b;
            b.x = sW[kk * 128 + t * 16 + lrow];
            b.y = sW[(kk + 1) * 128 + t * 16 + lrow];
            acc[t] = __builtin_amdgcn_wmma_f32_16x16x4_f32(
                false, a, false, b, (short)0, acc[t], false, false);
        }
    }

    // C/D layout: VGPR r -> row m0+r (lanes 0-15) / m0+8+r (lanes 16-31)
    const int rbase = m0 + ((lane >> 4) << 3);
#pragma unroll
    for (int t = 0; t < 8; ++t)
#pragma unroll
        for (int r = 0; r < 8; ++r)
            C[(size_t)(rbase + r) * 128 + t * 16 + lrow] = acc[t][r];
}

// ---------------------------------------------------------------------------
// Per-node attention logits: al[node*4+head] = dot(h[node, head*32:...], att)
// ---------------------------------------------------------------------------
__global__ __launch_bounds__(256) void logits_kernel(const float* __restrict__ h,
                                                     const float* __restrict__ att_src,
                                                     const float* __restrict__ att_dst,
                                                     float* __restrict__ al_src,
                                                     float* __restrict__ al_dst) {
    int t = blockIdx.x * 256 + threadIdx.x;
    if (t >= NNODES * HEADS) return;
    int node = t >> 2, head = t & 3;
    const float* hp = h + (size_t)node * 128 + head * 32;
    float s1 = 0.f, s2 = 0.f;
#pragma unroll
    for (int c = 0; c < 32; ++c) {
        float v = hp[c];
        s1 += v * att_src[head * 32 + c];
        s2 += v * att_dst[head * 32 + c];
    }
    al_src[t] = s1;
    al_dst[t] = s2;
}

__global__ __launch_bounds__(256) void init_kernel(float* __restrict__ m,
                                                   float* __restrict__ denom,
                                                   float* __restrict__ out) {
    int i = blockIdx.x * 256 + threadIdx.x;
    if (i < NNODES * HEADS) { m[i] = -INFINITY; denom[i] = 0.0f; }
    if (i < NNODES * 128) out[i] = 0.0f;
}

__device__ __forceinline__ void atomicMaxF32(float* addr, float val) {
    if (val >= 0.0f) atomicMax((int*)addr, __float_as_int(val));
    else             atomicMin((unsigned int*)addr, __float_as_uint(val));
}

__device__ __forceinline__ void edge_ids(int e, const int* __restrict__ ei, int& s, int& d) {
    if (e < NEDGES) { s = ei[e]; d = ei[NEDGES + e]; }
    else            { s = d = e - NEDGES; }           // self-loops
}

__global__ __launch_bounds__(256) void edge_max_kernel(const int* __restrict__ ei,
                                                       const float* __restrict__ al_src,
                                                       const float* __restrict__ al_dst,
                                                       float* __restrict__ m) {
    int e = blockIdx.x * 256 + threadIdx.x;
    if (e >= ETOT) return;
    int s, d; edge_ids(e, ei, s, d);
#pragma unroll
    for (int h = 0; h < HEADS; ++h) {
        float v = al_src[s * 4 + h] + al_dst[d * 4 + h];
        v = v > 0.0f ? v : NEGSLOPE * v;
        atomicMaxF32(&m[d * 4 + h], v);
    }
}

__global__ __launch_bounds__(256) void edge_sum_kernel(const int* __restrict__ ei,
                                                       const float* __restrict__ al_src,
                                                       const float* __restrict__ al_dst,
                                                       const float* __restrict__ m,
                                                       float* __restrict__ denom) {
    int e = blockIdx.x * 256 + threadIdx.x;
    if (e >= ETOT) return;
    int s, d; edge_ids(e, ei, s, d);
#pragma unroll
    for (int h = 0; h < HEADS; ++h) {
        float v = al_src[s * 4 + h] + al_dst[d * 4 + h];
        v = v > 0.0f ? v : NEGSLOPE * v;
        atomicAdd(&denom[d * 4 + h], __expf(v - m[d * 4 + h]));
    }
}

// One wave per edge; lane owns 4 consecutive channels (head = lane>>3).
__global__ __launch_bounds__(256) void edge_msg_kernel(const int* __restrict__ ei,
                                                       const float* __restrict__ h,
                                                       const float* __restrict__ al_src,
                                                       const float* __restrict__ al_dst,
                                                       const float* __restrict__ m,
                                                       const float* __restrict__ denom,
                                                       float* __restrict__ out) {
    int e = blockIdx.x * 8 + (threadIdx.x >> 5);
    if (e >= ETOT) return;
    const int lane = threadIdx.x & 31;
    int s, d; edge_ids(e, ei, s, d);

    const int head = lane >> 3;
    float v = al_src[s * 4 + head] + al_dst[d * 4 + head];
    v = v > 0.0f ? v : NEGSLOPE * v;
    const float alpha = __expf(v - m[d * 4 + head]) / (denom[d * 4 + head] + 1e-16f);

    const float4 hv = *(const float4*)(h + (size_t)s * 128 + lane * 4);
    float* op = out + (size_t)d * 128 + lane * 4;
    atomicAdd(op + 0, hv.x * alpha);
    atomicAdd(op + 1, hv.y * alpha);
    atomicAdd(op + 2, hv.z * alpha);
    atomicAdd(op + 3, hv.w * alpha);
}

// ---------------------------------------------------------------------------
extern "C" void kernel_launch(void* const* d_in, const int* in_sizes, int n_in,
                              void* d_out, int out_size, void* d_ws, size_t ws_size,
                              hipStream_t stream) {
    const float* x   = (const float*)d_in[0];
    const int*   ei  = (const int*)  d_in[1];
    const float* W1  = (const float*)d_in[2];
    const float* as1 = (const float*)d_in[3];
    const float* ad1 = (const float*)d_in[4];
    const float* W2  = (const float*)d_in[5];
    const float* as2 = (const float*)d_in[6];
    const float* ad2 = (const float*)d_in[7];
    float* out = (float*)d_out;

    float* ws   = (float*)d_ws;
    float* hbuf = ws;                                  // N*128
    float* out1 = hbuf + (size_t)NNODES * 128;         // N*128
    float* al_s = out1 + (size_t)NNODES * 128;         // N*4
    float* al_d = al_s + NNODES * HEADS;               // N*4
    float* mbuf = al_d + NNODES * HEADS;               // N*4
    float* dbuf = mbuf + NNODES * HEADS;               // N*4

    const dim3 blk(256);
    const int g_gemm  = (NNODES + 127) / 128;
    const int g_nh    = (NNODES * HEADS + 255) / 256;
    const int g_nfull = (NNODES * 128 + 255) / 256;
    const int g_edge  = (ETOT + 255) / 256;
    const int g_emsg  = (ETOT + 7) / 8;

    // ---- layer 1 ----
    gemm128_wmma<<<g_gemm, blk, 0, stream>>>(x, W1, hbuf, NNODES, 0);
    logits_kernel<<<g_nh, blk, 0, stream>>>(hbuf, as1, ad1, al_s, al_d);
    init_kernel<<<g_nfull, blk, 0, stream>>>(mbuf, dbuf, out1);
    edge_max_kernel<<<g_edge, blk, 0, stream>>>(ei, al_s, al_d, mbuf);
    edge_sum_kernel<<<g_edge, blk, 0, stream>>>(ei, al_s, al_d, mbuf, dbuf);
    edge_msg_kernel<<<g_emsg, blk, 0, stream>>>(ei, hbuf, al_s, al_d, mbuf, dbuf, out1);

    // ---- layer 2 (ReLU folded into GEMM A-load) ----
    gemm128_wmma<<<g_gemm, blk, 0, stream>>>(out1, W2, hbuf, NNODES, 1);
    logits_kernel<<<g_nh, blk, 0, stream>>>(hbuf, as2, ad2, al_s, al_d);
    init_kernel<<<g_nfull, blk, 0, stream>>>(mbuf, dbuf, out);
    edge_max_kernel<<<g_edge, blk, 0, stream>>>(ei, al_s, al_d, mbuf);
    edge_sum_kernel<<<g_edge, blk, 0, stream>>>(ei, al_s, al_d, mbuf, dbuf);
    edge_msg_kernel<<<g_emsg, blk, 0, stream>>>(ei, hbuf, al_s, al_d, mbuf, dbuf, out);
}